// Candemann_Parafac_module_79396765434100
// MI455X (gfx1250) — compile-verified
//
#include <hip/hip_runtime.h>

typedef __attribute__((ext_vector_type(2))) float v2f;
typedef __attribute__((ext_vector_type(8))) float v8f;

#define NGAMMA 7
#define NALPHA 64
#define KDIM   (NGAMMA * NALPHA)   // 448
#define CPRANK 64

// ---------------------------------------------------------------------------
// Stage 1: w[i][j] = sum_r gamma[r][i] * alpha[r][j]   (7x64 = 448 outputs)
// Tiny: one block, each thread does a 64-term dot product.
// ---------------------------------------------------------------------------
__global__ void cp_weight_kernel(const float* __restrict__ gamma,
                                 const float* __restrict__ alpha,
                                 float* __restrict__ w) {
  int t = threadIdx.x;
  if (t >= KDIM) return;
  int i = t / NALPHA;   // 0..6
  int j = t % NALPHA;   // 0..63
  float s = 0.0f;
#pragma unroll 8
  for (int r = 0; r < CPRANK; ++r)
    s = fmaf(gamma[r * NGAMMA + i], alpha[r * NALPHA + j], s);
  w[t] = s;
}

// ---------------------------------------------------------------------------
// Stage 2: streaming GEMV via V_WMMA_F32_16X16X4_F32.
// Each wave32 owns 16 samples. Per iteration it feeds an A fragment
// (16 rows x 4 k-values, fp32) and a B fragment whose value depends only on K
// (w[k..k+3] broadcast over all 16 columns), accumulating into C across
// 448/4 = 112 WMMAs (two interleaved chains for ILP).
//
// A layout (ISA 7.12.2, 32-bit A 16x4): lane = m + 16*h, VGPR0=K(2h),
// VGPR1=K(2h+1) -> one aligned b64 load per lane per step.
// D layout: VGPR r -> M=r (lanes 0-15), M=r+8 (lanes 16-31); since all
// columns of B are identical, every lane in a half holds the same 8 results,
// and lanes 0 / 16 write them out.
// ---------------------------------------------------------------------------
__global__ void __launch_bounds__(256)
cp_dot_kernel(const float* __restrict__ x, const float* __restrict__ w,
              const float* __restrict__ beta0, float* __restrict__ out,
              int ntiles) {
  __shared__ float ws[KDIM];
  for (int t = threadIdx.x; t < KDIM; t += 256) ws[t] = w[t];
  __syncthreads();

  int wave = (int)((blockIdx.x * 256u + threadIdx.x) >> 5);
  if (wave >= ntiles) return;          // wave-uniform: EXEC stays all-1s

  int lane = threadIdx.x & 31;
  int m = lane & 15;                   // sample row within tile
  int h = lane >> 4;                   // K half (0 -> K0/K1, 1 -> K2/K3)

  const float* xp = x + (size_t)(wave * 16 + m) * KDIM + 2 * h;
  const float* wp = ws + 2 * h;

  v8f c0 = {};
  v8f c1 = {};
#pragma unroll 4
  for (int k = 0; k < KDIM; k += 8) {
    v2f a0 = *(const v2f*)(xp + k);
    v2f b0 = {wp[k], wp[k + 1]};
    c0 = __builtin_amdgcn_wmma_f32_16x16x4_f32(false, a0, false, b0,
                                               (short)0, c0, false, false);
    v2f a1 = *(const v2f*)(xp + k + 4);
    v2f b1 = {wp[k + 4], wp[k + 5]};
    c1 = __builtin_amdgcn_wmma_f32_16x16x4_f32(false, a1, false, b1,
                                               (short)0, c1, false, false);
  }

  v8f c = c0 + c1;
  float beta = beta0[0];
  if (m == 0) {
    float* o = out + (size_t)wave * 16 + h * 8;
#pragma unroll
    for (int r = 0; r < 8; ++r) o[r] = c[r] + beta;
  }
}

// ---------------------------------------------------------------------------
// Harness entry. Inputs: d_in[0]=x [N,7,64] f32, d_in[1]=beta_0 (1 elem),
// d_in[2]=gamma [64,7], d_in[3]=alpha [64,64]. Output: y [N] f32.
// d_ws holds the 448-float w matrix between the two kernels.
// ---------------------------------------------------------------------------
extern "C" void kernel_launch(void* const* d_in, const int* in_sizes, int n_in,
                              void* d_out, int out_size, void* d_ws, size_t ws_size,
                              hipStream_t stream) {
  const float* x     = (const float*)d_in[0];
  const float* beta0 = (const float*)d_in[1];
  const float* gamma = (const float*)d_in[2];
  const float* alpha = (const float*)d_in[3];
  float* out = (float*)d_out;
  float* w   = (float*)d_ws;   // 448 floats of scratch

  int N = out_size;            // 131072 samples (multiple of 16)
  int ntiles = N / 16;         // 16 samples per wave

  cp_weight_kernel<<<1, 512, 0, stream>>>(gamma, alpha, w);

  const int wavesPerBlock = 8; // 256 threads = 8 wave32
  int blocks = (ntiles + wavesPerBlock - 1) / wavesPerBlock;
  cp_dot_kernel<<<blocks, 256, 0, stream>>>(x, w, beta0, out, ntiles);
}